// D_attention_962072674526
// MI455X (gfx1250) — compile-verified
//
#include <hip/hip_runtime.h>
#include <math.h>

#define NEG_SLOPE 0.01f
#define WSTRIDE 1026   // 16-row padded LDS stride (keeps 8B align, spreads banks)

typedef float v2f __attribute__((ext_vector_type(2)));
typedef float v8f __attribute__((ext_vector_type(8)));

// ---------------- K1: c[h] = x0 . W_i[h] + b[h] (one wave per head) -------
__global__ __launch_bounds__(256) void k_head_bias(const float* __restrict__ x,
                                                   const float* __restrict__ W,
                                                   const float* __restrict__ b,
                                                   float* __restrict__ c) {
    int lane = threadIdx.x & 31;
    int h = threadIdx.x >> 5;                 // 8 waves = 8 heads
    const float* wrow = W + (size_t)h * 2048; // W_i[h] = W[h, 0:1024]
    float acc = 0.f;
    for (int d = lane; d < 1024; d += 32) acc += x[d] * wrow[d];
    for (int off = 16; off; off >>= 1) acc += __shfl_xor(acc, off, 32);
    if (lane == 0) c[h] = acc + b[h];
}

// ---------------- K2: scores GEMM via V_WMMA_F32_16X16X4_F32 --------------
// Each wave computes a 16(row) x 16(head, 8 valid) tile of
//   s[n,h] = c[h] + x[n,:].W_j[h,:]   then leaky-ReLU, store, per-head max.
__global__ __launch_bounds__(256) void k_scores(const float* __restrict__ x,
                                                const float* __restrict__ W,
                                                const float* __restrict__ c,
                                                float* __restrict__ scores,
                                                float* __restrict__ pmax) {
    __shared__ float Wl[16 * WSTRIDE];   // W_j padded to 16 heads, ~64KB
    __shared__ float smax[64];
    const int tid = threadIdx.x;

    // Preload W_j (rows 8..15 zero) into LDS.
    for (int i = tid; i < 16 * 1024; i += 256) {
        int h = i >> 10, d = i & 1023;
        Wl[h * WSTRIDE + d] = (h < 8) ? W[(size_t)h * 2048 + 1024 + d] : 0.f;
    }
    __syncthreads();

    const int wv   = tid >> 5;
    const int lane = tid & 31;
    const int base = blockIdx.x * 128 + wv * 16;   // 16 rows per wave
    const int mrow = lane & 15;                    // A row / B column(head)
    const int kh   = lane >> 4;                    // K-half (ISA A/B layout)

    const float* xrow = x  + (size_t)(base + mrow) * 1024 + 2 * kh;
    const float* brow = Wl + mrow * WSTRIDE + 2 * kh;

    v8f acc = {};
    #pragma unroll 4
    for (int k = 0; k < 1024; k += 4) {
        v2f a  = *(const v2f*)(xrow + k);   // A: x[base+mrow, k+2kh .. +1]
        v2f bb = *(const v2f*)(brow + k);   // B: W_j[head,   k+2kh .. +1]
        acc = __builtin_amdgcn_wmma_f32_16x16x4_f32(
                  false, a, false, bb, (short)0, acc, false, false);
    }

    // Epilogue: bias + leaky ReLU + store + per-head tile max.
    const int head = mrow;
    float cmax = -INFINITY;
    if (head < 8) {
        float ch = c[head];
        #pragma unroll
        for (int r = 0; r < 8; ++r) {               // C: M = r + 8*kh
            int row = base + r + 8 * kh;
            float s = acc[r] + ch;
            s = (s >= 0.f) ? s : NEG_SLOPE * s;
            scores[(size_t)row * 8 + head] = s;
            cmax = fmaxf(cmax, s);
        }
    }
    cmax = fmaxf(cmax, __shfl_xor(cmax, 16, 32));   // merge the two K-halves
    if (lane < 8) smax[wv * 8 + lane] = cmax;
    __syncthreads();
    if (tid < 8) {
        float m = smax[tid];
        for (int w = 1; w < 8; ++w) m = fmaxf(m, smax[w * 8 + tid]);
        pmax[blockIdx.x * 8 + tid] = m;
    }
}

// ---------------- K3: global per-head max ---------------------------------
__global__ void k_reduce_max(const float* __restrict__ pmax,
                             float* __restrict__ m, int nb) {
    int t = threadIdx.x;
    if (t < 8) {
        float v = -INFINITY;
        for (int i = 0; i < nb; ++i) v = fmaxf(v, pmax[i * 8 + t]);
        m[t] = v;
    }
}

// ---------------- K4: partial sum of exp(s - m) per head ------------------
__global__ __launch_bounds__(256) void k_sumexp(const float* __restrict__ scores,
                                                const float* __restrict__ m,
                                                float* __restrict__ psum) {
    __shared__ float sm[256];
    int t = threadIdx.x;
    float mh = m[t & 7];                       // stride 256 preserves head id
    size_t base = (size_t)blockIdx.x * 1024;   // 128 rows * 8 heads
    float acc = 0.f;
    #pragma unroll
    for (int j = 0; j < 4; ++j) acc += expf(scores[base + t + 256 * j] - mh);
    sm[t] = acc;
    __syncthreads();
    if (t < 8) {
        float s = 0.f;
        for (int j = 0; j < 32; ++j) s += sm[t + 8 * j];
        psum[blockIdx.x * 8 + t] = s;
    }
}

// ---------------- K5: global per-head Z -----------------------------------
__global__ void k_reduce_sum(const float* __restrict__ psum,
                             float* __restrict__ Z, int nb) {
    int t = threadIdx.x;
    if (t < 8) {
        float s = 0.f;
        for (int i = 0; i < nb; ++i) s += psum[i * 8 + t];
        Z[t] = s;
    }
}

// ---------------- K6: per-row collapsed weight w[n] -----------------------
// w[n] = sum_h exp(s[n,h]-m[h]) / Z[h]   (b128 loads: 2 per row)
__global__ __launch_bounds__(256) void k_weights(const float* __restrict__ scores,
                                                 const float* __restrict__ m,
                                                 const float* __restrict__ Z,
                                                 float* __restrict__ wout) {
    __shared__ float mh[8], zi[8];
    if (threadIdx.x < 8) { mh[threadIdx.x] = m[threadIdx.x];
                           zi[threadIdx.x] = 1.f / Z[threadIdx.x]; }
    __syncthreads();
    int n = blockIdx.x * 256 + threadIdx.x;
    const float4* s4 = (const float4*)(scores + (size_t)n * 8);  // 16B aligned
    float4 lo = s4[0], hi = s4[1];
    float s = expf(lo.x - mh[0]) * zi[0] + expf(lo.y - mh[1]) * zi[1]
            + expf(lo.z - mh[2]) * zi[2] + expf(lo.w - mh[3]) * zi[3]
            + expf(hi.x - mh[4]) * zi[4] + expf(hi.y - mh[5]) * zi[5]
            + expf(hi.z - mh[6]) * zi[6] + expf(hi.w - mh[7]) * zi[7];
    wout[n] = s;
}

// ---------------- K7: partial out[d] = sum_n w[n]*x[n,d] ------------------
// Streaming pass over x (L2-resident after K2). Each thread owns 4
// consecutive d's -> one global_load_b128 per row per thread, fully coalesced.
__global__ __launch_bounds__(256) void k_wsum(const float* __restrict__ x,
                                              const float* __restrict__ w,
                                              float* __restrict__ pout) {
    int d0 = threadIdx.x * 4;                  // 256 threads * 4 = 1024 cols
    int n0 = blockIdx.x * 128;
    float a0 = 0.f, a1 = 0.f, a2 = 0.f, a3 = 0.f;
    for (int n = n0; n < n0 + 128; ++n) {
        float wn = w[n];                       // wave-uniform broadcast
        const float* xr = x + (size_t)n * 1024 + d0;
        __builtin_prefetch(x + (size_t)(n + 8) * 1024 + d0, 0, 3);
        float4 xv = *(const float4*)xr;        // global_load_b128
        a0 += wn * xv.x;
        a1 += wn * xv.y;
        a2 += wn * xv.z;
        a3 += wn * xv.w;
    }
    float4 r; r.x = a0; r.y = a1; r.z = a2; r.w = a3;
    *(float4*)(pout + (size_t)blockIdx.x * 1024 + d0) = r;
}

// ---------------- K8: reduce partials, relu(/H) ---------------------------
__global__ __launch_bounds__(256) void k_final(const float* __restrict__ pout,
                                               float* __restrict__ out, int nb) {
    int d0 = threadIdx.x * 4;                  // 1 block, 256 threads, 4 d each
    float a0 = 0.f, a1 = 0.f, a2 = 0.f, a3 = 0.f;
    for (int i = 0; i < nb; ++i) {
        float4 p = *(const float4*)(pout + (size_t)i * 1024 + d0);
        a0 += p.x; a1 += p.y; a2 += p.z; a3 += p.w;
    }
    float4 r;
    r.x = fmaxf(a0 * 0.125f, 0.f);
    r.y = fmaxf(a1 * 0.125f, 0.f);
    r.z = fmaxf(a2 * 0.125f, 0.f);
    r.w = fmaxf(a3 * 0.125f, 0.f);
    *(float4*)(out + d0) = r;
}

extern "C" void kernel_launch(void* const* d_in, const int* in_sizes, int n_in,
                              void* d_out, int out_size, void* d_ws, size_t ws_size,
                              hipStream_t stream) {
    (void)in_sizes; (void)n_in; (void)out_size; (void)ws_size;
    const float* x = (const float*)d_in[0];   // (16384, 1024) fp32
    const float* W = (const float*)d_in[1];   // (8, 2048)     fp32
    const float* b = (const float*)d_in[2];   // (8,)          fp32
    float* out = (float*)d_out;               // (1024,)       fp32
    float* ws  = (float*)d_ws;

    // Workspace layout (floats): ~1.1 MB total; all float4 regions 16B aligned.
    float* c      = ws;               // 8
    float* scores = c + 8;            // 16384*8   (offset 32B -> 16B aligned)
    float* pmax   = scores + 131072;  // 128*8
    float* m      = pmax + 1024;      // 8
    float* psum   = m + 8;            // 128*8
    float* Z      = psum + 1024;      // 8
    float* wrow   = Z + 8;            // 16384
    float* pout   = wrow + 16384;     // 128*1024  (offset 598112B -> 16B aligned)

    k_head_bias <<<  1, 256, 0, stream>>>(x, W, b, c);
    k_scores    <<<128, 256, 0, stream>>>(x, W, c, scores, pmax);
    k_reduce_max<<<  1,  32, 0, stream>>>(pmax, m, 128);
    k_sumexp    <<<128, 256, 0, stream>>>(scores, m, psum);
    k_reduce_sum<<<  1,  32, 0, stream>>>(psum, Z, 128);
    k_weights   <<< 64, 256, 0, stream>>>(scores, m, Z, wrow);
    k_wsum      <<<128, 256, 0, stream>>>(x, wrow, pout);
    k_final     <<<  1, 256, 0, stream>>>(pout, out, 128);
}